// CrossLayerLight_25220047962583
// MI455X (gfx1250) — compile-verified
//
#include <hip/hip_runtime.h>

typedef float v2f __attribute__((ext_vector_type(2)));
typedef float v8f __attribute__((ext_vector_type(8)));

static constexpr int BB  = 2;
static constexpr int NP  = 8192;
static constexpr int CH  = 64;
static constexpr int KNN = 16;

__device__ __forceinline__ float leaky(float x) { return fmaxf(x, 0.1f * x); }

// ---------------------------------------------------------------------------
// kNN: one thread per query, candidates tiled through LDS, register top-16.
// ---------------------------------------------------------------------------
__global__ __launch_bounds__(256) void knn_kernel(
    const float* __restrict__ xq,   // [B,3,N] queries
    const float* __restrict__ xk,   // [B,3,N] keys
    int* __restrict__ idx)          // [B,N,16]
{
    __shared__ float sx[256], sy[256], sz[256];
    int t = blockIdx.x * 256 + threadIdx.x;
    int b = t / NP;
    int n = t % NP;
    const float* q  = xq + (size_t)b * 3 * NP;
    const float* kp = xk + (size_t)b * 3 * NP;
    float qx = q[n], qy = q[NP + n], qz = q[2 * NP + n];

    float best[KNN]; int bidx[KNN];
#pragma unroll
    for (int j = 0; j < KNN; j++) { best[j] = 3.4e38f; bidx[j] = 0; }

    for (int tile = 0; tile < NP; tile += 256) {
        __syncthreads();
        int c = tile + threadIdx.x;
        sx[threadIdx.x] = kp[c];
        sy[threadIdx.x] = kp[NP + c];
        sz[threadIdx.x] = kp[2 * NP + c];
        __syncthreads();
        for (int j = 0; j < 256; j++) {
            float dx = qx - sx[j], dy = qy - sy[j], dz = qz - sz[j];
            float d  = dx * dx + dy * dy + dz * dz;
            if (d < best[KNN - 1]) {
                float vd = d; int vi = tile + j;
#pragma unroll
                for (int s = 0; s < KNN; s++) {
                    if (vd < best[s]) {
                        float td = best[s]; int ti = bidx[s];
                        best[s] = vd; bidx[s] = vi;
                        vd = td; vi = ti;
                    }
                }
            }
        }
    }
    int* o = idx + (size_t)t * KNN;
#pragma unroll
    for (int j = 0; j < KNN; j++) o[j] = bidx[j];
}

// ---------------------------------------------------------------------------
// Stage 64x64 weight W (out-major) into LDS as WMMA B fragments:
//   wl[(kk*4+nc)*32 + lane] = { W[o][k0], W[o][k0+1] },
//   o = nc*16 + lane%16, k0 = kk*4 + 2*(lane/16)
// ---------------------------------------------------------------------------
__device__ __forceinline__ void stage_wfrag(v2f* wl, const float* __restrict__ W,
                                            int tid, int nthreads)
{
    for (int s = tid; s < 2048; s += nthreads) {
        int lane = s & 31, nc = (s >> 5) & 3, kk = s >> 7;
        int o  = nc * 16 + (lane & 15);
        int k0 = kk * 4 + 2 * (lane >> 4);
        v2f w; w.x = W[o * 64 + k0]; w.y = W[o * 64 + k0 + 1];
        wl[s] = w;
    }
}

// ---------------------------------------------------------------------------
// GEMM, channel-major input: out_rm[b,n,m] = sum_c in[b,c,n]*W[m,c] + bias[m]
// One wave per 16-row tile; v_wmma_f32_16x16x4_f32.
// ---------------------------------------------------------------------------
__global__ __launch_bounds__(256) void gemm_cm_kernel(
    const float* __restrict__ in,    // [B,CH,N]
    const float* __restrict__ W,     // [64,64]
    const float* __restrict__ bias,  // [64]
    float* __restrict__ out_rm)      // [B,N,64]
{
    __shared__ v2f  wl[2048];
    __shared__ float bl[64];
    int tid = threadIdx.x;
    stage_wfrag(wl, W, tid, 256);
    if (tid < 64) bl[tid] = bias[tid];
    __syncthreads();

    int lane = tid & 31, wave = tid >> 5;
    int tile = blockIdx.x * 8 + wave;           // [0, B*N/16)
    int b  = tile / (NP / 16);
    int n0 = (tile % (NP / 16)) * 16;
    const float* inb = in + (size_t)b * CH * NP;
    int row   = n0 + (lane & 15);
    int khalf = 2 * (lane >> 4);

    v2f a[16];
#pragma unroll
    for (int kk = 0; kk < 16; kk++) {
        int k0 = kk * 4 + khalf;
        v2f v; v.x = inb[(size_t)k0 * NP + row]; v.y = inb[(size_t)(k0 + 1) * NP + row];
        a[kk] = v;
    }
#pragma unroll
    for (int nc = 0; nc < 4; nc++) {
        v8f acc = {0.f, 0.f, 0.f, 0.f, 0.f, 0.f, 0.f, 0.f};
#pragma unroll
        for (int kk = 0; kk < 16; kk++) {
            acc = __builtin_amdgcn_wmma_f32_16x16x4_f32(
                false, a[kk], false, wl[(kk * 4 + nc) * 32 + lane],
                (short)0, acc, false, false);
        }
        int m = nc * 16 + (lane & 15);
        float bv = bl[m];
        float* ob = out_rm + (size_t)b * NP * 64;
#pragma unroll
        for (int v = 0; v < 8; v++) {
            int r = n0 + v + 8 * (lane >> 4);
            ob[(size_t)r * 64 + m] = acc[v] + bv;
        }
    }
}

// ---------------------------------------------------------------------------
// GEMM, row-major input (trailing t1/t2): writes channel-major output and an
// optional row-major copy (feeds the final cross stage).
// ---------------------------------------------------------------------------
__global__ __launch_bounds__(256) void gemm_rm_kernel(
    const float* __restrict__ in,    // [B,N,64]
    const float* __restrict__ W,
    const float* __restrict__ bias,
    float* __restrict__ out_cm,      // [B,64,N]
    float* __restrict__ out_rm)      // [B,N,64] (nullable)
{
    __shared__ v2f  wl[2048];
    __shared__ float bl[64];
    int tid = threadIdx.x;
    stage_wfrag(wl, W, tid, 256);
    if (tid < 64) bl[tid] = bias[tid];
    __syncthreads();

    int lane = tid & 31, wave = tid >> 5;
    int tile = blockIdx.x * 8 + wave;
    int b  = tile / (NP / 16);
    int n0 = (tile % (NP / 16)) * 16;
    const float* inb = in + (size_t)b * NP * 64;
    int row   = n0 + (lane & 15);
    int khalf = 2 * (lane >> 4);

    v2f a[16];
#pragma unroll
    for (int kk = 0; kk < 16; kk++)
        a[kk] = *(const v2f*)(inb + (size_t)row * 64 + kk * 4 + khalf);

#pragma unroll
    for (int nc = 0; nc < 4; nc++) {
        v8f acc = {0.f, 0.f, 0.f, 0.f, 0.f, 0.f, 0.f, 0.f};
#pragma unroll
        for (int kk = 0; kk < 16; kk++) {
            acc = __builtin_amdgcn_wmma_f32_16x16x4_f32(
                false, a[kk], false, wl[(kk * 4 + nc) * 32 + lane],
                (short)0, acc, false, false);
        }
        int m = nc * 16 + (lane & 15);
        float bv = bl[m];
#pragma unroll
        for (int v = 0; v < 8; v++) {
            int r = n0 + v + 8 * (lane >> 4);
            float val = acc[v] + bv;
            out_cm[(size_t)b * 64 * NP + (size_t)m * NP + r] = val;
            if (out_rm) out_rm[((size_t)b * NP + r) * 64 + m] = val;
        }
    }
}

// ---------------------------------------------------------------------------
// Cross stage: one wave per query point. 16 neighbors == one WMMA M-tile.
// Phase 1: gather + pos-proj + leaky into LDS (16x64 block).
// Phase 2: per layer, 4 col-tiles x 16 k-steps of v_wmma_f32_16x16x4_f32;
//          last layer max-pools over the 16 rows (v8 reduce + xor-16 shuffle).
// ---------------------------------------------------------------------------
__global__ __launch_bounds__(128) void cross_kernel(
    const float* __restrict__ xq, const float* __restrict__ xk,   // [B,3,N]
    const float* __restrict__ p1, const float* __restrict__ p2,   // [B,N,64]
    const int*   __restrict__ idx,                                // [B,N,16]
    const float* __restrict__ posw, const float* __restrict__ posb,
    const float* __restrict__ w1, const float* __restrict__ b1,
    const float* __restrict__ w2, const float* __restrict__ b2,
    int nlayers,
    float* __restrict__ out_rm,   // [B,N,64]
    float* __restrict__ out_cm)   // [B,64,N] (nullable)
{
    __shared__ v2f  wl1[2048];
    __shared__ v2f  wl2[2048];
    __shared__ float bl1[64], bl2[64];
    __shared__ float pw[192], pb[64];
    __shared__ float X[4][16][68];   // per-wave 16x64 activation block (pitch 68)

    int tid = threadIdx.x;
    stage_wfrag(wl1, w1, tid, 128);
    if (nlayers > 1) stage_wfrag(wl2, w2, tid, 128);
    for (int s = tid; s < 192; s += 128) pw[s] = posw[s];
    if (tid < 64) {
        bl1[tid] = b1[tid];
        bl2[tid] = (nlayers > 1) ? b2[tid] : 0.f;
        pb[tid]  = posb[tid];
    }
    __syncthreads();

    int lane = tid & 31, wave = tid >> 5;
    int q = blockIdx.x * 4 + wave;          // [0, B*N)
    int b = q / NP;
    int n = q % NP;

    const float* xqb = xq + (size_t)b * 3 * NP;
    const float* xkb = xk + (size_t)b * 3 * NP;
    float qx = xqb[n], qy = xqb[NP + n], qz = xqb[2 * NP + n];   // wave-uniform
    const int* iq = idx + (size_t)q * KNN;                       // wave-uniform

    // Phase 1: build X[wave][r][c] = leaky(g2 + p1 + posf), lane owns 2 cols.
    int c0 = 2 * lane;
    v2f p1v = *(const v2f*)(p1 + (size_t)q * 64 + c0);
    float pwa0 = pw[c0 * 3 + 0], pwa1 = pw[c0 * 3 + 1], pwa2 = pw[c0 * 3 + 2];
    float pwb0 = pw[(c0 + 1) * 3 + 0], pwb1 = pw[(c0 + 1) * 3 + 1], pwb2 = pw[(c0 + 1) * 3 + 2];
    float pb0 = pb[c0], pb1 = pb[c0 + 1];
#pragma unroll
    for (int r = 0; r < KNN; r++) {
        int nb = iq[r];                                          // scalar load
        float dx = xkb[nb] - qx, dy = xkb[NP + nb] - qy, dz = xkb[2 * NP + nb] - qz;
        v2f g = *(const v2f*)(p2 + ((size_t)b * NP + nb) * 64 + c0);
        X[wave][r][c0]     = leaky(g.x + p1v.x + pb0 + dx * pwa0 + dy * pwa1 + dz * pwa2);
        X[wave][r][c0 + 1] = leaky(g.y + p1v.y + pb1 + dx * pwb0 + dy * pwb1 + dz * pwb2);
    }
    __syncthreads();

    // Phase 2: MLP layers via WMMA.
    int mrow  = lane & 15;
    int khalf = 2 * (lane >> 4);
    for (int layer = 0; layer < nlayers; layer++) {
        const v2f*  wl = layer ? wl2 : wl1;
        const float* bl = layer ? bl2 : bl1;
        bool last = (layer == nlayers - 1);

        v2f a[16];
#pragma unroll
        for (int kk = 0; kk < 16; kk++)
            a[kk] = *(const v2f*)&X[wave][mrow][kk * 4 + khalf];

#pragma unroll
        for (int nc = 0; nc < 4; nc++) {
            v8f acc = {0.f, 0.f, 0.f, 0.f, 0.f, 0.f, 0.f, 0.f};
#pragma unroll
            for (int kk = 0; kk < 16; kk++) {
                acc = __builtin_amdgcn_wmma_f32_16x16x4_f32(
                    false, a[kk], false, wl[(kk * 4 + nc) * 32 + lane],
                    (short)0, acc, false, false);
            }
            int m = nc * 16 + mrow;
            float bv = bl[m];
            if (!last) {
#pragma unroll
                for (int v = 0; v < 8; v++)
                    X[wave][v + 8 * (lane >> 4)][m] = leaky(acc[v] + bv);
            } else {
                float mx = -3.4e38f;
#pragma unroll
                for (int v = 0; v < 8; v++) mx = fmaxf(mx, leaky(acc[v] + bv));
                mx = fmaxf(mx, __shfl_xor(mx, 16, 32));   // combine M halves
                if (lane < 16) {
                    out_rm[(size_t)q * 64 + m] = mx;
                    if (out_cm)
                        out_cm[(size_t)b * 64 * NP + (size_t)m * NP + n] = mx;
                }
            }
        }
        __syncthreads();   // X rewritten between layers (uniform barrier count)
    }
}

// ---------------------------------------------------------------------------
extern "C" void kernel_launch(void* const* d_in, const int* in_sizes, int n_in,
                              void* d_out, int out_size, void* d_ws, size_t ws_size,
                              hipStream_t stream)
{
    (void)in_sizes; (void)n_in; (void)out_size; (void)ws_size;
    const float* pc1   = (const float*)d_in[0];
    const float* pc2   = (const float*)d_in[1];
    const float* feat1 = (const float*)d_in[2];
    const float* feat2 = (const float*)d_in[3];
    const float* t11_w = (const float*)d_in[4];  const float* t11_b = (const float*)d_in[5];
    const float* t22_w = (const float*)d_in[6];  const float* t22_b = (const float*)d_in[7];
    const float* pos1w = (const float*)d_in[8];  const float* pos1b = (const float*)d_in[9];
    const float* m1w1  = (const float*)d_in[10]; const float* m1b1  = (const float*)d_in[11];
    const float* m1w2  = (const float*)d_in[12]; const float* m1b2  = (const float*)d_in[13];
    const float* t1_w  = (const float*)d_in[14]; const float* t1_b  = (const float*)d_in[15];
    const float* t2_w  = (const float*)d_in[16]; const float* t2_b  = (const float*)d_in[17];
    const float* pos2w = (const float*)d_in[18]; const float* pos2b = (const float*)d_in[19];
    const float* m2w1  = (const float*)d_in[20]; const float* m2b1  = (const float*)d_in[21];

    float* out = (float*)d_out;
    const size_t S = (size_t)BB * NP * 64;     // 1,048,576 floats per [B,64,N] tensor
    float* ws = (float*)d_ws;
    float* q1 = ws;           float* q2 = ws + S;
    float* r1 = ws + 2 * S;   float* r2 = ws + 3 * S;
    float* pooled1 = ws + 4 * S;
    float* pooled2 = ws + 5 * S;
    int* idx1 = (int*)(ws + 6 * S);
    int* idx2 = idx1 + (size_t)BB * NP * KNN;
    float* fn1_rm  = q1;   // safe reuse: transforms done before trailing GEMMs
    float* fn2_rm  = q2;
    float* pooled3 = r1;

    // kNN (idx for cross3 == idx1, since cross3 reuses (pc1, pc2))
    knn_kernel<<<BB * NP / 256, 256, 0, stream>>>(pc1, pc2, idx1);
    knn_kernel<<<BB * NP / 256, 256, 0, stream>>>(pc2, pc1, idx2);

    // q1=f1@t11, q2=f2@t22, r1=f2@t11, r2=f1@t22
    gemm_cm_kernel<<<128, 256, 0, stream>>>(feat1, t11_w, t11_b, q1);
    gemm_cm_kernel<<<128, 256, 0, stream>>>(feat2, t22_w, t22_b, q2);
    gemm_cm_kernel<<<128, 256, 0, stream>>>(feat2, t11_w, t11_b, r1);
    gemm_cm_kernel<<<128, 256, 0, stream>>>(feat1, t22_w, t22_b, r2);

    // cross #1, #2 (mlp1: two layers)
    cross_kernel<<<BB * NP / 4, 128, 0, stream>>>(pc1, pc2, q1, q2, idx1,
        pos1w, pos1b, m1w1, m1b1, m1w2, m1b2, 2, pooled1, nullptr);
    cross_kernel<<<BB * NP / 4, 128, 0, stream>>>(pc2, pc1, r1, r2, idx2,
        pos1w, pos1b, m1w1, m1b1, m1w2, m1b2, 2, pooled2, nullptr);

    // trailing transforms -> fn1, fn2 (both output layout and row-major feed)
    gemm_rm_kernel<<<128, 256, 0, stream>>>(pooled1, t1_w, t1_b, out,       fn1_rm);
    gemm_rm_kernel<<<128, 256, 0, stream>>>(pooled2, t2_w, t2_b, out + S,   fn2_rm);

    // cross #3 (mlp2: one layer) -> ff written channel-major directly
    cross_kernel<<<BB * NP / 4, 128, 0, stream>>>(pc1, pc2, fn1_rm, fn2_rm, idx1,
        pos2w, pos2b, m2w1, m2b1, nullptr, nullptr, 1, pooled3, out + 2 * S);
}